// GCNEncoder_66211215835753
// MI455X (gfx1250) — compile-verified
//
#include <hip/hip_runtime.h>
#include <hip/hip_bf16.h>

typedef float v2f __attribute__((ext_vector_type(2)));
typedef float v8f __attribute__((ext_vector_type(8)));

// ---------------------------------------------------------------------------
// Degree / normalization kernels
// ---------------------------------------------------------------------------
__global__ void k_init_deg(float* __restrict__ deg, int n) {
    int i = blockIdx.x * blockDim.x + threadIdx.x;
    if (i < n) deg[i] = 1.0f;                 // self-loop contributes 1
}

__global__ void k_count_deg(float* __restrict__ deg, const int* __restrict__ dst, int e) {
    int i = blockIdx.x * blockDim.x + threadIdx.x;
    if (i < e) atomicAdd(&deg[dst[i]], 1.0f);
}

__global__ void k_finalize_dinv(float* __restrict__ deg, int n) {
    int i = blockIdx.x * blockDim.x + threadIdx.x;
    if (i < n) deg[i] = rsqrtf(deg[i]);       // deg >= 1 always (self-loops)
}

// ---------------------------------------------------------------------------
// Fully-fp32 GEMM: C[nrows x 64] = A[nrows x 64] * B[64 x 64]
// One wave per 16x16 output tile, V_WMMA_F32_16X16X4_F32, K-loop 16 steps.
// Block = 128 threads = 4 waves = the 4 column tiles of one 16-row stripe.
// ---------------------------------------------------------------------------
__global__ __launch_bounds__(128) void k_gemm64_wmma(const float* __restrict__ A,
                                                     const float* __restrict__ B,
                                                     float* __restrict__ C,
                                                     int nrows) {
    const int wave = threadIdx.x >> 5;        // 0..3 -> column tile
    const int lane = threadIdx.x & 31;
    const int half = lane >> 4;               // 0: K pair {0,1}, 1: K pair {2,3}
    const int ml   = lane & 15;
    const int row0 = blockIdx.x * 16;
    const int n    = wave * 16 + ml;          // output column for B/C/D lanes

    // clamp defensively (nrows is a multiple of 16 in practice, no divergence)
    int arow = row0 + ml;
    if (arow >= nrows) arow = nrows - 1;
    const float* __restrict__ ap = A + (size_t)arow * 64;

    v8f acc = {};
#pragma unroll
    for (int k = 0; k < 64; k += 4) {
        const int ka = k + half * 2;
        // A 16x4 tile: lane holds K={ka, ka+1} of row (row0+ml)
        v2f a = *(const v2f*)(ap + ka);
        // B 4x16 tile: lane holds rows K={ka, ka+1} at column n
        v2f b;
        b.x = B[(size_t)ka * 64 + n];
        b.y = B[(size_t)(ka + 1) * 64 + n];
        acc = __builtin_amdgcn_wmma_f32_16x16x4_f32(
            /*neg_a=*/false, a, /*neg_b=*/false, b,
            /*c_mod=*/(short)0, acc, /*reuse_a=*/false, /*reuse_b=*/false);
    }

    // D layout: VGPR j -> row (j + half*8), column n
#pragma unroll
    for (int j = 0; j < 8; ++j) {
        const int m = row0 + j + half * 8;
        if (m < nrows) C[(size_t)m * 64 + n] = acc[j];
    }
}

// ---------------------------------------------------------------------------
// out[i,d] = h[i,d]*dinv[i]^2 + bias[d]   (self-loop term + bias, full init)
// ---------------------------------------------------------------------------
__global__ void k_agg_init(float* __restrict__ out, const float* __restrict__ h,
                           const float* __restrict__ dinv, const float* __restrict__ bias,
                           int n) {
    int idx = blockIdx.x * blockDim.x + threadIdx.x;
    if (idx >= n * 64) return;
    int i = idx >> 6;
    int d = idx & 63;
    float di = dinv[i];
    out[idx] = h[idx] * (di * di) + bias[d];
}

// ---------------------------------------------------------------------------
// Edge scatter-add: one wave per edge, 2 features per lane (coalesced atomics)
// ---------------------------------------------------------------------------
__global__ void k_agg_edges(float* __restrict__ out, const float* __restrict__ h,
                            const float* __restrict__ dinv,
                            const int* __restrict__ src, const int* __restrict__ dst,
                            int e) {
    long long t = (long long)blockIdx.x * blockDim.x + threadIdx.x;
    int ed = (int)(t >> 5);
    if (ed >= e) return;
    int lane = (int)(t & 31);
    int s = src[ed];
    int d = dst[ed];
    float coef = dinv[s] * dinv[d];
    const float* __restrict__ hs = h + (size_t)s * 64;
    float* __restrict__ od = out + (size_t)d * 64;
    atomicAdd(&od[lane],      hs[lane]      * coef);
    atomicAdd(&od[lane + 32], hs[lane + 32] * coef);
}

// ---------------------------------------------------------------------------
// act = (u > 0.5) ? 2*max(v,0) : 0      (ReLU + inverted dropout, p=0.5)
// ---------------------------------------------------------------------------
__global__ void k_relu_dropout(float* __restrict__ act, const float* __restrict__ v,
                               const float* __restrict__ u, int total) {
    int idx = blockIdx.x * blockDim.x + threadIdx.x;
    if (idx >= total) return;
    float x = v[idx];
    x = fmaxf(x, 0.0f);
    act[idx] = (u[idx] > 0.5f) ? (x * 2.0f) : 0.0f;
}

// ---------------------------------------------------------------------------
// Orchestration
// ---------------------------------------------------------------------------
extern "C" void kernel_launch(void* const* d_in, const int* in_sizes, int n_in,
                              void* d_out, int out_size, void* d_ws, size_t ws_size,
                              hipStream_t stream) {
    const float* x  = (const float*)d_in[0];
    const float* W1 = (const float*)d_in[1];
    const float* b1 = (const float*)d_in[2];
    const float* W2 = (const float*)d_in[3];
    const float* b2 = (const float*)d_in[4];
    const float* u  = (const float*)d_in[5];
    const int*   ei = (const int*)d_in[6];

    const int N = in_sizes[0] / 64;
    const int E = in_sizes[6] / 2;
    const int* src = ei;
    const int* dst = ei + E;

    float* ws   = (float*)d_ws;
    float* dinv = ws;                            // N floats
    float* bufA = ws + N;                        // N*64 floats
    float* bufB = bufA + (size_t)N * 64;         // N*64 floats
    float* out  = (float*)d_out;

    const int total = N * 64;
    const int TB = 256;

    // normalization coefficients (shared by both layers; same graph)
    k_init_deg<<<(N + TB - 1) / TB, TB, 0, stream>>>(dinv, N);
    k_count_deg<<<(E + TB - 1) / TB, TB, 0, stream>>>(dinv, dst, E);
    k_finalize_dinv<<<(N + TB - 1) / TB, TB, 0, stream>>>(dinv, N);

    const int gemmBlocks = (N + 15) / 16;
    const int edgeBlocks = (int)(((long long)E * 32 + TB - 1) / TB);

    // ---- layer 1: h1 = x @ W1 ; agg(+b1) ; relu+dropout ----
    k_gemm64_wmma<<<gemmBlocks, 128, 0, stream>>>(x, W1, bufA, N);
    k_agg_init<<<(total + TB - 1) / TB, TB, 0, stream>>>(bufB, bufA, dinv, b1, N);
    k_agg_edges<<<edgeBlocks, TB, 0, stream>>>(bufB, bufA, dinv, src, dst, E);
    k_relu_dropout<<<(total + TB - 1) / TB, TB, 0, stream>>>(bufA, bufB, u, total);

    // ---- layer 2: h2 = act @ W2 ; agg(+b2) -> d_out ----
    k_gemm64_wmma<<<gemmBlocks, 128, 0, stream>>>(bufA, W2, bufB, N);
    k_agg_init<<<(total + TB - 1) / TB, TB, 0, stream>>>(out, bufB, dinv, b2, N);
    k_agg_edges<<<edgeBlocks, TB, 0, stream>>>(out, bufB, dinv, src, dst, E);
}